// oraclKalmanFilt_paper_delayed_39178691674382
// MI455X (gfx1250) — compile-verified
//
#include <hip/hip_runtime.h>
#include <hip/hip_bf16.h>

typedef __attribute__((ext_vector_type(2))) float v2f;
typedef __attribute__((ext_vector_type(4))) float v4f;
typedef __attribute__((ext_vector_type(8))) float v8f;

// ---------------------------------------------------------------------------
// Tiled FP32 WMMA GEMM:  C[M,N] = A[M,K] * op(B),  op = B (NN) or B^T (NT)
// Block tile 64x128, K staged in 16-wide LDS tiles, 8 waves (wave32).
// Wave (wr, wc): rows [wr*16, +16), cols [wc*64, +64) -> four v8f accumulators.
// B is stored k-pair-packed in LDS so both A and B fragments are contiguous
// 8-byte loads. All fragments for a K-tile are loaded into registers first,
// then 16 WMMAs issue back-to-back (single ds wait per tile).
// ---------------------------------------------------------------------------
template <bool TB>
__global__ __launch_bounds__(256) void gemm_wmma_k(const float* __restrict__ A,
                                                   const float* __restrict__ B,
                                                   float* __restrict__ C,
                                                   int M, int Nc, int Kc) {
  __shared__ float As[64][20];    // 64 rows x 16 k (+pad)
  __shared__ float Bs[8][260];    // k-pair p = k>>1; element (k,n) = Bs[p][n*2+(k&1)]

  const int tid  = threadIdx.x;
  const int lane = tid & 31;
  const int wave = tid >> 5;
  const int wr   = wave >> 1;     // 0..3  (16 rows each)
  const int wc   = wave & 1;      // 0..1  (64 cols each)
  const int br   = blockIdx.y * 64;
  const int bc   = blockIdx.x * 128;

  v8f acc0 = {}, acc1 = {}, acc2 = {}, acc3 = {};

  const int am = tid >> 2;          // 0..63  A-tile row
  const int ak = (tid & 3) << 2;    // 0,4,8,12

  const int m  = lane & 15;
  const int kp = (lane >> 4) << 1;  // 0 or 2

  for (int k0 = 0; k0 < Kc; k0 += 16) {
    // stage A tile (64x16), coalesced float4 loads
    {
      v4f av = *(const v4f*)(A + (size_t)(br + am) * Kc + k0 + ak);
      As[am][ak + 0] = av.x; As[am][ak + 1] = av.y;
      As[am][ak + 2] = av.z; As[am][ak + 3] = av.w;
    }
    // stage op(B) tile (16 k x 128 n), k-pair packed
    if (!TB) {
      const int bk = tid >> 4;            // 0..15
      const int bn = (tid & 15) << 3;     // 0..120
      const float* src = B + (size_t)(k0 + bk) * Nc + bc + bn;
      v4f b0 = *(const v4f*)(src);
      v4f b1 = *(const v4f*)(src + 4);
      const int p = bk >> 1, h = bk & 1;
      Bs[p][(bn + 0) * 2 + h] = b0.x; Bs[p][(bn + 1) * 2 + h] = b0.y;
      Bs[p][(bn + 2) * 2 + h] = b0.z; Bs[p][(bn + 3) * 2 + h] = b0.w;
      Bs[p][(bn + 4) * 2 + h] = b1.x; Bs[p][(bn + 5) * 2 + h] = b1.y;
      Bs[p][(bn + 6) * 2 + h] = b1.z; Bs[p][(bn + 7) * 2 + h] = b1.w;
    } else {
      const int bn = tid >> 1;            // 0..127 (row of B = col of op(B))
      const int bk = (tid & 1) << 3;      // 0 or 8
      const float* src = B + (size_t)(bc + bn) * Kc + k0 + bk;
      v4f b0 = *(const v4f*)(src);
      v4f b1 = *(const v4f*)(src + 4);
      const int p = bk >> 1;
      *(v2f*)&Bs[p + 0][bn * 2] = (v2f){b0.x, b0.y};
      *(v2f*)&Bs[p + 1][bn * 2] = (v2f){b0.z, b0.w};
      *(v2f*)&Bs[p + 2][bn * 2] = (v2f){b1.x, b1.y};
      *(v2f*)&Bs[p + 3][bn * 2] = (v2f){b1.z, b1.w};
    }
    __syncthreads();

    // Load ALL fragments for this K-tile into registers first...
    v2f af[4];
    v2f bf[4][4];
#pragma unroll
    for (int ks = 0; ks < 4; ++ks) {
      const int kb = ks << 2;
      af[ks] = *(const v2f*)&As[wr * 16 + m][kb + kp];
      const v2f* bp = (const v2f*)&Bs[(kb + kp) >> 1][(wc * 64 + m) * 2];
      bf[ks][0] = bp[0];
      bf[ks][1] = bp[16];
      bf[ks][2] = bp[32];
      bf[ks][3] = bp[48];
    }
    // ...then issue 16 WMMAs back-to-back on the matrix pipe
#pragma unroll
    for (int ks = 0; ks < 4; ++ks) {
      acc0 = __builtin_amdgcn_wmma_f32_16x16x4_f32(false, af[ks], false,
                                                   bf[ks][0], (short)0, acc0,
                                                   false, false);
      acc1 = __builtin_amdgcn_wmma_f32_16x16x4_f32(false, af[ks], false,
                                                   bf[ks][1], (short)0, acc1,
                                                   false, false);
      acc2 = __builtin_amdgcn_wmma_f32_16x16x4_f32(false, af[ks], false,
                                                   bf[ks][2], (short)0, acc2,
                                                   false, false);
      acc3 = __builtin_amdgcn_wmma_f32_16x16x4_f32(false, af[ks], false,
                                                   bf[ks][3], (short)0, acc3,
                                                   false, false);
    }
    __syncthreads();
  }

  const int cn = lane & 15;
  const int mg = (lane >> 4) << 3;   // 0 or 8
#pragma unroll
  for (int r = 0; r < 8; ++r) {
    const size_t row = (size_t)(br + wr * 16 + mg + r);
    float* cp = C + row * Nc + bc + wc * 64 + cn;
    cp[0]  = acc0[r];
    cp[16] = acc1[r];
    cp[32] = acc2[r];
    cp[48] = acc3[r];
  }
}

// ---------------------------------------------------------------------------
// Small helper kernels
// ---------------------------------------------------------------------------
__global__ void fill_k(float* p, float v, size_t n) {
  size_t i = (size_t)blockIdx.x * blockDim.x + threadIdx.x;
  if (i < n) p[i] = v;
}

// out = alpha*A + beta*B (elementwise)
__global__ void ew2_k(float* out, const float* A, const float* B, float alpha,
                      float beta, size_t n) {
  size_t i = (size_t)blockIdx.x * blockDim.x + threadIdx.x;
  if (i < n) out[i] = alpha * A[i] + beta * B[i];
}

// out = alpha*A + diag*I  (n x n)
__global__ void ewI_k(float* out, const float* A, float alpha, float dg, int n) {
  size_t i = (size_t)blockIdx.x * blockDim.x + threadIdx.x;
  if (i >= (size_t)n * n) return;
  int r = (int)(i / n), c = (int)(i % n);
  out[i] = alpha * A[i] + (r == c ? dg : 0.0f);
}

// D1 = a3*L + a2*I ; D0 = a3*L^2 + a2*L + a1*I   (coeffs read on-device)
__global__ void polymats_k(float* D0, float* D1, const float* L,
                           const float* LSQ, const float* a, int n) {
  size_t i = (size_t)blockIdx.x * blockDim.x + threadIdx.x;
  if (i >= (size_t)n * n) return;
  int r = (int)(i / n), c = (int)(i % n);
  float a1 = a[1], a2 = a[2], a3 = a[3];
  D1[i] = a3 * L[i] + (r == c ? a2 : 0.0f);
  D0[i] = a3 * LSQ[i] + a2 * L[i] + (r == c ? a1 : 0.0f);
}

__global__ void gemv_k(const float* A, const float* x, float* y, int M, int K) {
  int i = blockIdx.x * blockDim.x + threadIdx.x;
  if (i >= M) return;
  const float* row = A + (size_t)i * K;
  float s = 0.0f;
  for (int k = 0; k < K; ++k) s += row[k] * x[k];
  y[i] = s;
}

// y = A*x + a[idx]*q
__global__ void gemv_axpy_k(const float* A, const float* x, const float* q,
                            const float* a, int idx, float* y, int M, int K) {
  int i = blockIdx.x * blockDim.x + threadIdx.x;
  if (i >= M) return;
  const float* row = A + (size_t)i * K;
  float s = 0.0f;
  for (int k = 0; k < K; ++k) s += row[k] * x[k];
  y[i] = s + a[idx] * q[i];
}

__global__ void vscale_dev_k(float* out, const float* x, const float* a,
                             int idx, int n) {
  int i = blockIdx.x * blockDim.x + threadIdx.x;
  if (i < n) out[i] = a[idx] * x[i];
}

// L += s_e * (e_n - e_m)(e_n - e_m)^T  for each edge
__global__ void scatterL_k(float* L, const int* edges, const float* sv, int E,
                           int N) {
  int e = blockIdx.x * blockDim.x + threadIdx.x;
  if (e >= E) return;
  int n = edges[e], m = edges[E + e];
  float v = sv[e];
  atomicAdd(&L[(size_t)n * N + n], v);
  atomicAdd(&L[(size_t)m * N + m], v);
  atomicAdd(&L[(size_t)n * N + m], -v);
  atomicAdd(&L[(size_t)m * N + n], -v);
}

// diff_c[p,e] = c_p[n_e] - c_p[m_e],  p = 0..2
__global__ void diffc_k(float* dc, const float* c0, const float* c1,
                        const float* c2, const int* edges, int E) {
  int e = blockIdx.x * blockDim.x + threadIdx.x;
  if (e >= E) return;
  int n = edges[e], m = edges[E + e];
  dc[e]         = c0[n] - c0[m];
  dc[E + e]     = c1[n] - c1[m];
  dc[2 * E + e] = c2[n] - c2[m];
}

// H[r,e] = sum_p dc[p,e]*(D_p[r,n_e]-D_p[r,m_e]); D2 = a3*I, D3 = 0
__global__ void h_k(float* H, const float* D0, const float* D1, const float* dc,
                    const int* edges, const float* a, int N, int E) {
  size_t idx = (size_t)blockIdx.x * blockDim.x + threadIdx.x;
  if (idx >= (size_t)N * E) return;
  int r = (int)(idx / E), e = (int)(idx % E);
  int n = edges[e], m = edges[E + e];
  float a3 = a[3];
  float v = dc[e] * (D0[(size_t)r * N + n] - D0[(size_t)r * N + m]) +
            dc[E + e] * (D1[(size_t)r * N + n] - D1[(size_t)r * N + m]) +
            dc[2 * E + e] * a3 *
                ((r == n ? 1.0f : 0.0f) - (r == m ? 1.0f : 0.0f));
  H[idx] = v;
}

// H2[r,j] = H[r, conn[j]] computed directly
__global__ void h2_k(float* H2, const float* D0, const float* D1,
                     const float* dc, const int* edges, const int* conn,
                     const float* a, int N, int E, int C) {
  size_t idx = (size_t)blockIdx.x * blockDim.x + threadIdx.x;
  if (idx >= (size_t)N * C) return;
  int r = (int)(idx / C), j = (int)(idx % C);
  int e = conn[j];
  int n = edges[e], m = edges[E + e];
  float a3 = a[3];
  float v = dc[e] * (D0[(size_t)r * N + n] - D0[(size_t)r * N + m]) +
            dc[E + e] * (D1[(size_t)r * N + n] - D1[(size_t)r * N + m]) +
            dc[2 * E + e] * a3 *
                ((r == n ? 1.0f : 0.0f) - (r == m ? 1.0f : 0.0f));
  H2[idx] = v;
}

__global__ void rowsum_abs_k(const float* S, float* rs, int n) {
  int i = blockIdx.x * blockDim.x + threadIdx.x;
  if (i >= n) return;
  const float* row = S + (size_t)i * n;
  float s = 0.0f;
  for (int k = 0; k < n; ++k) s += fabsf(row[k]);
  rs[i] = s;
}

__global__ void maxred_k(const float* v, float* out, int n) {
  __shared__ float sm[256];
  float m = 0.0f;
  for (int i = threadIdx.x; i < n; i += 256) m = fmaxf(m, v[i]);
  sm[threadIdx.x] = m;
  __syncthreads();
  for (int s = 128; s > 0; s >>= 1) {
    if ((int)threadIdx.x < s)
      sm[threadIdx.x] = fmaxf(sm[threadIdx.x], sm[threadIdx.x + s]);
    __syncthreads();
  }
  if (threadIdx.x == 0) out[0] = sm[0];
}

// X0 = (1/||S||_inf) * I
__global__ void initinv_k(float* X, const float* mp, int n) {
  size_t i = (size_t)blockIdx.x * blockDim.x + threadIdx.x;
  if (i >= (size_t)n * n) return;
  int r = (int)(i / n), c = (int)(i % n);
  X[i] = (r == c) ? (1.0f / mp[0]) : 0.0f;
}

__global__ void scatter_s_k(float* outs, const float* s1, const int* conn,
                            int C) {
  int j = blockIdx.x * blockDim.x + threadIdx.x;
  if (j >= C) return;
  int e = conn[j];
  outs[e] = s1[e];
}

__global__ void gather_scc_k(float* SCC, const float* SIG, const int* conn,
                             int C, int E) {
  size_t idx = (size_t)blockIdx.x * blockDim.x + threadIdx.x;
  if (idx >= (size_t)C * C) return;
  int i = (int)(idx / C), j = (int)(idx % C);
  SCC[idx] = SIG[(size_t)conn[i] * E + conn[j]];
}

__global__ void scatter_sig_k(float* outsig, const float* SS, const int* conn,
                              int C, int E) {
  size_t idx = (size_t)blockIdx.x * blockDim.x + threadIdx.x;
  if (idx >= (size_t)C * C) return;
  int i = (int)(idx / C), j = (int)(idx % C);
  outsig[(size_t)conn[i] * E + conn[j]] = SS[idx];
}

// ---------------------------------------------------------------------------
// Host orchestration
// ---------------------------------------------------------------------------
extern "C" void kernel_launch(void* const* d_in, const int* in_sizes, int n_in,
                              void* d_out, int out_size, void* d_ws,
                              size_t ws_size, hipStream_t stream) {
  (void)n_in; (void)out_size; (void)ws_size;
  const float* F     = (const float*)d_in[0];
  const float* V     = (const float*)d_in[2];
  const float* W     = (const float*)d_in[3];
  const float* Sig0  = (const float*)d_in[4];
  const float* s_in  = (const float*)d_in[5];
  const float* a     = (const float*)d_in[6];
  const float* q     = (const float*)d_in[7];
  const float* y     = (const float*)d_in[8];
  const int*   edges = (const int*)d_in[9];
  const int*   conn  = (const int*)d_in[10];

  const int E = in_sizes[5];   // 2048
  const int N = in_sizes[7];   // 1024
  const int C = in_sizes[10];  // 1024
  const int NS_ITERS = 24;

  float* w = (float*)d_ws;
  size_t off = 0;
  auto alloc = [&](size_t nf) {
    float* p = w + off;
    off += (nf + 63) & ~(size_t)63;
    return p;
  };
  // E x E
  float* SIG  = alloc((size_t)E * E);
  float* TEE  = alloc((size_t)E * E);
  float* TEE2 = alloc((size_t)E * E);
  // N x E / E x N
  float* H    = alloc((size_t)N * E);
  float* TNE  = alloc((size_t)N * E);
  float* Kb   = alloc((size_t)E * N);
  // N x N pool
  float* L    = alloc((size_t)N * N);
  float* LSQ  = alloc((size_t)N * N);
  float* D0   = alloc((size_t)N * N);
  float* D1   = alloc((size_t)N * N);
  float* S    = alloc((size_t)N * N);
  float* SINV = alloc((size_t)N * N);
  float* Yb   = alloc((size_t)N * N);
  float* Ub   = alloc((size_t)N * N);
  // stage 2
  float* SCC  = alloc((size_t)C * C);
  float* H2   = alloc((size_t)N * C);
  float* P1   = alloc((size_t)C * C);
  float* P2   = alloc((size_t)C * C);
  float* P3   = alloc((size_t)C * C);
  // vectors
  float* s1    = alloc(E);
  float* c1    = alloc(N);
  float* c2    = alloc(N);
  float* pr    = alloc(N);
  float* pr2   = alloc(N);
  float* resid = alloc(N);
  float* dsv   = alloc(E);
  float* dc    = alloc((size_t)3 * E);
  float* rs    = alloc(N);
  float* rsmax = alloc(64);

  float* out_s   = (float*)d_out;
  float* out_sig = out_s + E;

  auto blk = [](size_t n) { return (unsigned)((n + 255) / 256); };
  auto gemm = [&](const float* A, const float* B, float* Cm, int M, int Nc,
                  int Kc, bool tb) {
    dim3 g(Nc / 128, M / 64), b(256);
    if (tb)
      gemm_wmma_k<true><<<g, b, 0, stream>>>(A, B, Cm, M, Nc, Kc);
    else
      gemm_wmma_k<false><<<g, b, 0, stream>>>(A, B, Cm, M, Nc, Kc);
  };
  auto invert_ns = [&](const float* Sp, float* X, int n) {
    rowsum_abs_k<<<blk(n), 256, 0, stream>>>(Sp, rs, n);
    maxred_k<<<1, 256, 0, stream>>>(rs, rsmax, n);
    initinv_k<<<blk((size_t)n * n), 256, 0, stream>>>(X, rsmax, n);
    for (int it = 0; it < NS_ITERS; ++it) {
      gemm(Sp, X, Yb, n, n, n, false);       // Y = S*X
      gemm(X, Yb, Ub, n, n, n, false);       // U = X*S*X
      ew2_k<<<blk((size_t)n * n), 256, 0, stream>>>(X, X, Ub, 2.0f, -1.0f,
                                                    (size_t)n * n);
    }
  };
  auto build_jacobian = [&](const float* sv) {
    fill_k<<<blk((size_t)N * N), 256, 0, stream>>>(L, 0.0f, (size_t)N * N);
    scatterL_k<<<blk(E), 256, 0, stream>>>(L, edges, sv, E, N);
    gemm(L, L, LSQ, N, N, N, false);
    polymats_k<<<blk((size_t)N * N), 256, 0, stream>>>(D0, D1, L, LSQ, a, N);
    gemv_k<<<blk(N), 256, 0, stream>>>(L, q, c1, N, N);
    gemv_k<<<blk(N), 256, 0, stream>>>(L, c1, c2, N, N);
    diffc_k<<<blk(E), 256, 0, stream>>>(dc, q, c1, c2, edges, E);
  };

  // ---- Stage 1: predict ----
  gemv_k<<<blk(E), 256, 0, stream>>>(F, s_in, s1, E, E);        // s = F s
  gemm(F, Sig0, TEE, E, E, E, false);                           // T = F*Sigma
  gemm(TEE, F, SIG, E, E, E, true);                             // SIG = T*F^T
  ew2_k<<<blk((size_t)E * E), 256, 0, stream>>>(SIG, SIG, V, 1.0f, 1.0f,
                                                (size_t)E * E); // + V

  // ---- L, H ----
  build_jacobian(s1);
  h_k<<<blk((size_t)N * E), 256, 0, stream>>>(H, D0, D1, dc, edges, a, N, E);

  // ---- S = H*SIG*H^T + W,  K = SIG*H^T*inv(S) ----
  gemm(H, SIG, TNE, N, E, E, false);                            // T1 = H*SIG
  gemm(TNE, H, S, N, N, E, true);                               // S = T1*H^T
  ew2_k<<<blk((size_t)N * N), 256, 0, stream>>>(S, S, W, 1.0f, 1.0f,
                                                (size_t)N * N); // + W
  invert_ns(S, SINV, N);
  gemm(SIG, H, TNE, E, N, E, true);                             // A2 = SIG*H^T
  gemm(TNE, SINV, Kb, E, N, N, false);                          // K = A2*Sinv

  // ---- state update: s += K*(y - poly(L,q,a)) ----
  vscale_dev_k<<<blk(N), 256, 0, stream>>>(pr, q, a, 3, N);     // r = a3*q
  gemv_axpy_k<<<blk(N), 256, 0, stream>>>(L, pr, q, a, 2, pr2, N, N);
  gemv_axpy_k<<<blk(N), 256, 0, stream>>>(L, pr2, q, a, 1, pr, N, N);
  gemv_axpy_k<<<blk(N), 256, 0, stream>>>(L, pr, q, a, 0, pr2, N, N);
  ew2_k<<<blk(N), 256, 0, stream>>>(resid, pr2, y, -1.0f, 1.0f, (size_t)N);
  gemv_k<<<blk(E), 256, 0, stream>>>(Kb, resid, dsv, E, N);
  ew2_k<<<blk(E), 256, 0, stream>>>(s1, s1, dsv, 1.0f, 1.0f, (size_t)E);

  // ---- covariance update: SIG = (I-KH)*SIG*(I-KH)^T + K*W*K^T ----
  gemm(Kb, H, TEE2, E, E, N, false);                            // KH
  ewI_k<<<blk((size_t)E * E), 256, 0, stream>>>(TEE2, TEE2, -1.0f, 1.0f, E);
  gemm(TEE2, SIG, TEE, E, E, E, false);                         // (I-KH)*SIG
  gemm(TEE, TEE2, SIG, E, E, E, true);                          // * (I-KH)^T
  gemm(Kb, W, TNE, E, N, N, false);                             // K*W
  gemm(TNE, Kb, TEE, E, E, N, true);                            // K*W*K^T
  ew2_k<<<blk((size_t)E * E), 256, 0, stream>>>(SIG, SIG, TEE, 1.0f, 1.0f,
                                                (size_t)E * E);

  // ---- mask s -> final s output ----
  fill_k<<<blk(E), 256, 0, stream>>>(out_s, 0.0f, (size_t)E);
  scatter_s_k<<<blk(C), 256, 0, stream>>>(out_s, s1, conn, C);

  // ---- Stage 2 on connections ----
  build_jacobian(out_s);  // L2, D0, D1, diffc from masked s
  h2_k<<<blk((size_t)N * C), 256, 0, stream>>>(H2, D0, D1, dc, edges, conn, a,
                                               N, E, C);
  gather_scc_k<<<blk((size_t)C * C), 256, 0, stream>>>(SCC, SIG, conn, C, E);

  gemm(H2, SCC, TNE, N, C, C, false);                           // H2*Scc
  gemm(TNE, H2, S, N, N, C, true);                              // S2
  ew2_k<<<blk((size_t)N * N), 256, 0, stream>>>(S, S, W, 1.0f, 1.0f,
                                                (size_t)N * N);
  invert_ns(S, SINV, N);
  gemm(SCC, H2, TNE, C, N, C, true);                            // Scc*H2^T
  gemm(TNE, SINV, Kb, C, N, N, false);                          // K2

  gemm(Kb, H2, P1, C, C, N, false);                             // K2*H2
  ewI_k<<<blk((size_t)C * C), 256, 0, stream>>>(P1, P1, -1.0f, 1.0f, C);
  gemm(P1, SCC, P2, C, C, C, false);                            // (I-K2H2)*Scc
  gemm(P2, P1, P3, C, C, C, true);                              // *(I-K2H2)^T
  gemm(Kb, W, TNE, C, N, N, false);                             // K2*W
  gemm(TNE, Kb, P2, C, C, N, true);                             // K2*W*K2^T
  ew2_k<<<blk((size_t)C * C), 256, 0, stream>>>(P3, P3, P2, 1.0f, 1.0f,
                                                (size_t)C * C);

  // ---- scatter Sigma_small into Sigma_out ----
  fill_k<<<blk((size_t)E * E), 256, 0, stream>>>(out_sig, 0.0f, (size_t)E * E);
  scatter_sig_k<<<blk((size_t)C * C), 256, 0, stream>>>(out_sig, P3, conn, C,
                                                        E);
}